// LogEig_28140625723420
// MI455X (gfx1250) — compile-verified
//
#include <hip/hip_runtime.h>

// Batched SPD matrix logarithm (B=8192, N=64) via inverse scaling-and-squaring:
//   A0 = X/trace(X);  k coupled Newton-Schulz sqrts;  log(I+E) Horner series;
//   log X = 2^k * log(A0^(1/2^k)) + log(t) I.
// All heavy math is 64x64 GEMMs in LDS using V_WMMA_F32_16X16X4_F32 (wave32).

typedef __attribute__((ext_vector_type(2))) float v2f;
typedef __attribute__((ext_vector_type(8))) float v8f;

#define NMAT   64
#define LD     68          // padded leading dim: stride-68 dwords -> conflict-free column fetch
#define LEVELS 5           // 2^5 = 32
#define TPB    256         // 8 wave32 per block, one block per matrix

// C = scale * (A @ B) + diagAdd * I    (64x64 in LDS, leading dim LD)
// Wave w owns output tiles (tm0, tn) and (tm0+32, tn) with tn=(w&3)*16,
// tm0=(w>>2)*16 -> both tiles share B fragments; two independent accumulator
// chains so consecutive v_wmma don't RAW on the same C/D registers.
// entry_sync: barrier making prior LDS writes visible (skip when caller knows
// operands were already fenced).
__device__ __forceinline__ void gemm64(float* __restrict__ C,
                                       const float* __restrict__ A,
                                       const float* __restrict__ B,
                                       float scale, float diagAdd,
                                       int lane, int wave, bool entry_sync) {
  if (entry_sync) __syncthreads();
  const int l15  = lane & 15;
  const int koff = (lane >> 4) << 1;          // 0 | 2
  const int tn   = (wave & 3) << 4;
  const int tm0  = (wave >> 2) << 4;
  const int tm1  = tm0 + 32;
  const int row0 = tm0 + l15;
  const int row1 = tm1 + l15;
  const int col  = tn + l15;
  v8f acc0 = {}, acc1 = {};
  #pragma unroll
  for (int k0 = 0; k0 < NMAT; k0 += 4) {
    v2f b, a0, a1;
    b.x  = B[(k0 + koff) * LD + col];          // shared B fragment
    b.y  = B[(k0 + koff + 1) * LD + col];
    a0.x = A[row0 * LD + k0 + koff];
    a0.y = A[row0 * LD + k0 + koff + 1];
    a1.x = A[row1 * LD + k0 + koff];
    a1.y = A[row1 * LD + k0 + koff + 1];
    acc0 = __builtin_amdgcn_wmma_f32_16x16x4_f32(false, a0, false, b,
                                                 (short)0, acc0, false, false);
    acc1 = __builtin_amdgcn_wmma_f32_16x16x4_f32(false, a1, false, b,
                                                 (short)0, acc1, false, false);
  }
  const int mb0 = tm0 + ((lane >> 4) << 3);
  const int mb1 = tm1 + ((lane >> 4) << 3);
  #pragma unroll
  for (int r = 0; r < 8; ++r) {
    float v0 = acc0[r] * scale;
    float v1 = acc1[r] * scale;
    if (mb0 + r == col) v0 += diagAdd;         // fused alpha*AB + beta*I
    if (mb1 + r == col) v1 += diagAdd;
    C[(mb0 + r) * LD + col] = v0;
    C[(mb1 + r) * LD + col] = v1;
  }
}

__global__ __launch_bounds__(TPB)
void logm_spd_kernel(const float* __restrict__ Xg, float* __restrict__ Og) {
  __shared__ float sm[5 * NMAT * LD];          // 5 ping-pong 64x68 f32 (~85 KB)
  __shared__ float sred[NMAT];
  __shared__ float s_logt, s_invt;

  float* Y = sm + 0 * NMAT * LD;               // sqrt chain
  float* Z = sm + 1 * NMAT * LD;               // inverse-sqrt chain
  float* P = sm + 2 * NMAT * LD;               // NS polynomial / Horner G
  float* W = sm + 3 * NMAT * LD;               // Y ping-pong / E = M - I
  float* V = sm + 4 * NMAT * LD;               // Z ping-pong / Horner G'

  const int tid  = threadIdx.x;
  const int lane = tid & 31;
  const int wave = tid >> 5;
  const long long base = (long long)blockIdx.x * (NMAT * NMAT);

  // ---- t = trace(X) -------------------------------------------------------
  if (tid < NMAT) sred[tid] = Xg[base + tid * NMAT + tid];
  __syncthreads();
  if (tid == 0) {
    float t = 0.0f;
    for (int i = 0; i < NMAT; ++i) t += sred[i];
    s_invt = 1.0f / t;
    s_logt = logf(t);
  }
  __syncthreads();
  const float invt = s_invt;

  // ---- Y = X/t (b128 loads), Z = I ---------------------------------------
  const float4* Xg4 = reinterpret_cast<const float4*>(Xg + base);
  #pragma unroll
  for (int v = 0; v < 4; ++v) {
    const int i4 = tid + v * TPB;              // coalesced float4 index
    float4 x = Xg4[i4];
    const int i = i4 << 2, r = i >> 6, c = i & 63;
    x.x *= invt; x.y *= invt; x.z *= invt; x.w *= invt;
    *reinterpret_cast<float4*>(&Y[r * LD + c]) = x;   // 16B-aligned (LD%4==0)
  }
  for (int i = tid; i < NMAT * NMAT; i += TPB) {
    const int r = i >> 6, c = i & 63;
    Z[r * LD + c] = (r == c) ? 1.0f : 0.0f;
  }

  // ---- LEVELS coupled Newton-Schulz square roots --------------------------
  #pragma unroll 1
  for (int lvl = 0; lvl < LEVELS; ++lvl) {
    if (lvl > 0) {
      __syncthreads();                         // prior gemm readers of Z done
      for (int i = tid; i < NMAT * NMAT; i += TPB) {
        const int r = i >> 6, c = i & 63;
        Z[r * LD + c] = (r == c) ? 1.0f : 0.0f;
      }
    }
    const int iters = (lvl == 0) ? 14 : (lvl == 1) ? 9 : (lvl == 2) ? 7
                    : (lvl == 3) ? 5 : 4;
    #pragma unroll 1
    for (int it = 0; it < iters; ++it) {
      gemm64(P, Z, Y, -0.5f, 1.5f, lane, wave, true);   // P = 1.5I - 0.5 Z@Y
      gemm64(W, Y, P,  1.0f, 0.0f, lane, wave, true);   // W = Y@P   (next Y)
      gemm64(V, P, Z,  1.0f, 0.0f, lane, wave, false);  // V = P@Z   (independent)
      float* s;
      s = Y; Y = W; W = s;
      s = Z; Z = V; V = s;
    }
  }

  // ---- E = M - I ----------------------------------------------------------
  __syncthreads();
  for (int i = tid; i < NMAT * NMAT; i += TPB) {
    const int r = i >> 6, c = i & 63;
    W[r * LD + c] = Y[r * LD + c] - ((r == c) ? 1.0f : 0.0f);
  }

  // ---- log(I+E) = E*(c1 I + c2 E + ... + c8 E^7), Horner ------------------
  float* G  = P;
  float* Gn = V;
  for (int i = tid; i < NMAT * NMAT; i += TPB) {
    const int r = i >> 6, c = i & 63;
    G[r * LD + c] = (r == c) ? (-1.0f / 8.0f) : 0.0f;   // c8
  }
  #pragma unroll 1
  for (int m = 7; m >= 1; --m) {
    const float cm = ((m & 1) ? 1.0f : -1.0f) / (float)m;
    gemm64(Gn, W, G, 1.0f, cm, lane, wave, true);       // G' = E@G + cm I
    float* s = G; G = Gn; Gn = s;
  }
  const float s2k = (float)(1 << LEVELS);
  gemm64(Z, W, G, s2k, 0.0f, lane, wave, true);         // L = 2^k * E@P(E)

  // ---- out = L + log(t) I  (b128 stores) ----------------------------------
  __syncthreads();
  const float logt = s_logt;
  float4* Og4 = reinterpret_cast<float4*>(Og + base);
  #pragma unroll
  for (int v = 0; v < 4; ++v) {
    const int i4 = tid + v * TPB;
    const int i = i4 << 2, r = i >> 6, c = i & 63;
    float4 o = *reinterpret_cast<const float4*>(&Z[r * LD + c]);
    if (r == c + 0) o.x += logt;
    if (r == c + 1) o.y += logt;
    if (r == c + 2) o.z += logt;
    if (r == c + 3) o.w += logt;
    Og4[i4] = o;
  }
}

extern "C" void kernel_launch(void* const* d_in, const int* in_sizes, int n_in,
                              void* d_out, int out_size, void* d_ws, size_t ws_size,
                              hipStream_t stream) {
  const float* X = (const float*)d_in[0];
  float* out = (float*)d_out;
  const int batch = in_sizes[0] / (NMAT * NMAT);        // 8192
  logm_spd_kernel<<<dim3(batch), dim3(TPB), 0, stream>>>(X, out);
}